// NetVLADLayer_33432025432607
// MI455X (gfx1250) — compile-verified
//
#include <hip/hip_runtime.h>
#include <math.h>

// ---------------------------------------------------------------------------
// NetVLAD for MI455X (gfx1250, wave32).
// Memory-bound op (~275MB traffic @23.3TB/s >> 5 GFLOP compute), so both
// GEMMs use fp32 WMMA (V_WMMA_F32_16X16X4_F32) to match the fp32 reference
// exactly in precision while still running on the matrix pipe.
// ---------------------------------------------------------------------------

typedef __attribute__((ext_vector_type(2))) float v2f;
typedef __attribute__((ext_vector_type(8))) float v8f;

#define NIMG 32
#define CDIM 128
#define KCLU 64
#define SPIX 4800   // 60*80
#define EPSN 1e-12f

// ---------------------------------------------------------------------------
// Kernel 0: per-pixel inverse L2 norm over channels.
// rnorm[n*S + s] = 1 / max(||x[n,:,s]||, eps)
// ---------------------------------------------------------------------------
__global__ __launch_bounds__(256) void k_rnorm(const float* __restrict__ x,
                                               float* __restrict__ rnorm) {
    int idx = blockIdx.x * 256 + threadIdx.x;           // n*S + s
    if (idx >= NIMG * SPIX) return;
    int n = idx / SPIX, s = idx - n * SPIX;
    const float* xp = x + (size_t)n * CDIM * SPIX + s;  // stride S over c, coalesced over s
    float ss = 0.f;
#pragma unroll 8
    for (int c = 0; c < CDIM; ++c) {
        float v = xp[(size_t)c * SPIX];
        ss += v * v;
    }
    rnorm[idx] = 1.0f / fmaxf(sqrtf(ss), EPSN);
}

// ---------------------------------------------------------------------------
// Kernel 1: logits = W @ xf  (per image), softmax over k, write a[N,K,S].
// Block = 128 threads (4 waves). Wave w computes the 16x16 tile
// k in [16w,16w+16) x s in [s0,s0+16), reducing over C in steps of 4 with
// V_WMMA_F32_16X16X4_F32.
//
// fp32 WMMA operand layout (ISA 7.12.2):
//   A (16x4, MxKd): lane l<16 -> M=l, VGPR0=Kd0,VGPR1=Kd1 ; lane>=16 -> Kd2,Kd3
//   B (4x16, KdxN): lane l<16 -> N=l, VGPR0=Kd0,VGPR1=Kd1 ; lane>=16 -> Kd2,Kd3
//   C/D (16x16):    VGPR i -> M=i+8*(l>>4), N=l&15
// ---------------------------------------------------------------------------
__global__ __launch_bounds__(128) void k_logits_softmax(const float* __restrict__ x,
                                                        const float* __restrict__ w,
                                                        const float* __restrict__ rnorm,
                                                        float* __restrict__ a) {
    __shared__ float lds_logit[KCLU * 16];   // [k][s_local]
    __shared__ float lds_max[16], lds_rsum[16];

    const int n    = blockIdx.y;
    const int s0   = blockIdx.x * 16;
    const int wave = threadIdx.x >> 5;
    const int lane = threadIdx.x & 31;
    const int l15  = lane & 15;
    const int half = lane >> 4;
    const int k0   = wave * 16;

    const float* xn = x + (size_t)n * CDIM * SPIX;
    const float  rn = rnorm[n * SPIX + s0 + l15];              // pixel for B lanes
    const float* wp = w + (k0 + l15) * CDIM + 2 * half;        // A row, Kd pair base
    const float* xp = xn + (size_t)(2 * half) * SPIX + s0 + l15;

    v8f acc = {};
#pragma unroll 4
    for (int cb = 0; cb < CDIM; cb += 4) {
        v2f av = *(const v2f*)(wp + cb);                       // w[k, cb+2h+{0,1}]
        v2f bv;
        bv.x = xp[(size_t)cb * SPIX] * rn;                     // xf[cb+2h,   s]
        bv.y = xp[(size_t)(cb + 1) * SPIX] * rn;               // xf[cb+2h+1, s]
        acc = __builtin_amdgcn_wmma_f32_16x16x4_f32(false, av, false, bv,
                                                    (short)0, acc, false, false);
    }

    // spill tile to LDS as [k][s_local]
#pragma unroll
    for (int i = 0; i < 8; ++i)
        lds_logit[(k0 + i + 8 * half) * 16 + l15] = acc[i];
    __syncthreads();

    // softmax stats over 64 clusters per pixel
    if (threadIdx.x < 16) {
        const int p = threadIdx.x;
        float m = -INFINITY;
        for (int k = 0; k < KCLU; ++k) m = fmaxf(m, lds_logit[k * 16 + p]);
        float ssum = 0.f;
        for (int k = 0; k < KCLU; ++k) ssum += __expf(lds_logit[k * 16 + p] - m);
        lds_max[p]  = m;
        lds_rsum[p] = 1.0f / ssum;
    }
    __syncthreads();

    // write a[n,k,s0+p]
    float* an = a + (size_t)n * KCLU * SPIX + s0;
    for (int idx = threadIdx.x; idx < KCLU * 16; idx += 128) {
        int k = idx >> 4, p = idx & 15;
        an[(size_t)k * SPIX + p] = __expf(lds_logit[idx] - lds_max[p]) * lds_rsum[p];
    }
}

// ---------------------------------------------------------------------------
// Kernel 2: vlad[n,k,c] = sum_s a[k,s]*xf[c,s] - (sum_s a[k,s])*cent[k,c].
// One wave per 16x16 (k,c) tile, reducing over S=4800 in steps of 4.
// A and B per-lane fetches are contiguous 64-bit pairs (s, s+1).
// Row sums of A accumulated alongside and folded via LDS.
// Writes directly into d_out (vlad layout == output layout pre-normalization).
// ---------------------------------------------------------------------------
__global__ __launch_bounds__(32) void k_vlad(const float* __restrict__ x,
                                             const float* __restrict__ a,
                                             const float* __restrict__ rnorm,
                                             const float* __restrict__ cent,
                                             float* __restrict__ out) {
    __shared__ float lds_ps[32];

    const int n    = blockIdx.z;
    const int k0   = blockIdx.y * 16;
    const int c0   = blockIdx.x * 16;
    const int lane = threadIdx.x;
    const int l15  = lane & 15;
    const int half = lane >> 4;

    const float* ap = a + (size_t)n * KCLU * SPIX + (size_t)(k0 + l15) * SPIX + 2 * half;
    const float* xp = x + ((size_t)n * CDIM + (c0 + l15)) * SPIX + 2 * half;
    const float* rp = rnorm + (size_t)n * SPIX + 2 * half;

    v8f   acc  = {};
    float psum = 0.f;
#pragma unroll 2
    for (int s = 0; s < SPIX; s += 4) {
        v2f av = *(const v2f*)(ap + s);     // a[k, s+2h+{0,1}]
        v2f bx = *(const v2f*)(xp + s);     // x[c, s+2h+{0,1}]
        v2f rv = *(const v2f*)(rp + s);     // rnorm[s+2h+{0,1}]
        v2f bv; bv.x = bx.x * rv.x; bv.y = bx.y * rv.y;
        psum += av.x + av.y;
        acc = __builtin_amdgcn_wmma_f32_16x16x4_f32(false, av, false, bv,
                                                    (short)0, acc, false, false);
    }

    // asum[k_local] = lds_ps[k_local] + lds_ps[k_local+16] (two s-phases)
    lds_ps[lane] = psum;
    __syncthreads();

    float*       op = out + ((size_t)n * KCLU + k0) * CDIM + c0 + l15;
    const float* cp = cent + (size_t)k0 * CDIM + c0 + l15;
#pragma unroll
    for (int i = 0; i < 8; ++i) {
        int   kl   = i + 8 * half;                  // C/D layout: M = i + 8*half
        float asum = lds_ps[kl] + lds_ps[kl + 16];
        op[(size_t)kl * CDIM] = acc[i] - asum * cp[(size_t)kl * CDIM];
    }
}

// ---------------------------------------------------------------------------
// Kernel 3: in-place intra-normalize over C then global normalize over K*C.
// One block per image; 4 lanes per (k) row, 32 values per lane in registers.
// ---------------------------------------------------------------------------
__global__ __launch_bounds__(256) void k_norm(float* __restrict__ out) {
    __shared__ float lds[8];
    __shared__ float gscale;

    float* p  = out + (size_t)blockIdx.x * KCLU * CDIM;
    const int t   = threadIdx.x;
    const int row = t >> 2;      // 0..63
    const int j   = t & 3;       // 4 lanes per row

    float vals[32];
    float ss = 0.f;
#pragma unroll
    for (int i = 0; i < 32; ++i) {
        vals[i] = p[row * CDIM + j * 32 + i];
        ss += vals[i] * vals[i];
    }
    // reduce across the 4 adjacent lanes of this row
    ss += __shfl_xor(ss, 1, 32);
    ss += __shfl_xor(ss, 2, 32);
    const float rrow = 1.0f / fmaxf(sqrtf(ss), EPSN);

    float gs = 0.f;
#pragma unroll
    for (int i = 0; i < 32; ++i) {
        vals[i] *= rrow;
        gs += vals[i] * vals[i];
    }
    // block-wide reduce of gs (wave32 shuffle tree + LDS)
    for (int off = 1; off < 32; off <<= 1) gs += __shfl_xor(gs, off, 32);
    if ((t & 31) == 0) lds[t >> 5] = gs;
    __syncthreads();
    if (t == 0) {
        float tot = 0.f;
        for (int i = 0; i < 8; ++i) tot += lds[i];
        gscale = 1.0f / fmaxf(sqrtf(tot), EPSN);
    }
    __syncthreads();
    const float sc = gscale;
#pragma unroll
    for (int i = 0; i < 32; ++i) p[row * CDIM + j * 32 + i] = vals[i] * sc;
}

// ---------------------------------------------------------------------------
extern "C" void kernel_launch(void* const* d_in, const int* in_sizes, int n_in,
                              void* d_out, int out_size, void* d_ws, size_t ws_size,
                              hipStream_t stream) {
    const float* x    = (const float*)d_in[0];   // [32,128,60,80]
    const float* w    = (const float*)d_in[1];   // [64,128]
    const float* cent = (const float*)d_in[2];   // [64,128]
    float*       out  = (float*)d_out;           // [32, 64*128]

    float* rnorm = (float*)d_ws;                 // N*S floats        (0.6 MB)
    float* a     = rnorm + NIMG * SPIX;          // N*K*S floats      (39.3 MB)

    k_rnorm<<<(NIMG * SPIX + 255) / 256, 256, 0, stream>>>(x, rnorm);
    k_logits_softmax<<<dim3(SPIX / 16, NIMG), 128, 0, stream>>>(x, w, rnorm, a);
    k_vlad<<<dim3(CDIM / 16, KCLU / 16, NIMG), 32, 0, stream>>>(x, a, rnorm, cent, out);
    k_norm<<<NIMG, 256, 0, stream>>>(out);
}